// FrontierPolicyNetwork_49065706390002
// MI455X (gfx1250) — compile-verified
//
#include <hip/hip_runtime.h>
#include <hip/hip_bf16.h>
#include <stdint.h>

// Problem constants (match reference)
#define NN   100000   // nodes
#define EE   600000   // edges
#define CC   8192     // candidates
#define BB   16       // batch
#define DIN  64
#define HH   128
#define NLAYER 3
#define PP   262144   // pool entries

typedef __attribute__((ext_vector_type(2))) float v2f;
typedef __attribute__((ext_vector_type(8))) float v8f;

__device__ __forceinline__ v8f v8f_zero() {
    v8f z;
    #pragma unroll
    for (int i = 0; i < 8; ++i) z[i] = 0.f;
    return z;
}

__device__ __forceinline__ v8f wmma4(v2f a, v2f b, v8f c) {
    // D = A(16x4 f32) * B(4x16 f32) + C(16x16 f32)
    return __builtin_amdgcn_wmma_f32_16x16x4_f32(
        /*neg_a=*/false, a, /*neg_b=*/false, b,
        /*c_mod=*/(short)0, c, /*reuse_a=*/false, /*reuse_b=*/false);
}

// ---------------------------------------------------------------------------
// Generic WMMA GEMM: out[M, 128] = act(in[M, K] @ W[K, 128] + bias
//                                      (+ extra[extraIdx[row], 128]))
// 256 threads = 8 waves; each wave computes a 16x128 output tile.
//
// LDS weight image is K-PAIR INTERLEAVED:
//   sW[p*256 + 2*col + s] = W[2p + s][col]   (p = k/2 pair index, s = parity)
// so a B fragment {W[kk][col], W[kk+1][col]} is ONE aligned ds_load_b64.
//
// A fragment (16x4 f32):   lane<16 -> row=lane,    holds K = k, k+1
//                          lane>=16-> row=lane-16, holds K = k+2, k+3
// C/D (16x16 f32, 8 VGPR): elem[v] = out[row0 + v + 8*(lane>>4)][16n + (lane&15)]
// ---------------------------------------------------------------------------
template <int K, bool RELU, bool EXTRA>
__global__ __launch_bounds__(256)
void k_gemm(const float* __restrict__ in, const float* __restrict__ W,
            const float* __restrict__ bias, float* __restrict__ out, int M,
            const float* __restrict__ extra, const int* __restrict__ extraIdx) {
    __shared__ float sW[K * HH];                 // up to 64 KB, interleaved
    // cooperative pair-interleaving stage: 2x float2 global -> 1x float4 LDS
    for (int u = threadIdx.x; u < (K / 2) * 64; u += 256) {
        const int p = u >> 6;                    // k-pair index
        const int c = (u & 63) * 2;              // column base (even)
        float2 r0 = *(const float2*)(W + (size_t)(2 * p) * HH + c);
        float2 r1 = *(const float2*)(W + (size_t)(2 * p + 1) * HH + c);
        float4 w;
        w.x = r0.x; w.y = r1.x; w.z = r0.y; w.w = r1.y;
        *(float4*)(sW + p * 256 + c * 2) = w;
    }
    __syncthreads();

    const int lane = threadIdx.x & 31;
    const int wave = threadIdx.x >> 5;
    const int half = lane >> 4;
    const int l16  = lane & 15;
    const int row0 = blockIdx.x * 128 + wave * 16;

    int arow = row0 + l16;
    if (arow > M - 1) arow = M - 1;              // clamp: keeps EXEC all-ones
    const float* __restrict__ ap = in + (size_t)arow * K;

    v8f acc[8];
    #pragma unroll
    for (int n = 0; n < 8; ++n) acc[n] = v8f_zero();

    for (int k = 0; k < K; k += 8) {             // two WMMA k-steps per iter
        __builtin_prefetch(ap + k + 64, 0, 1);   // global_prefetch_b8
        v2f a0 = *(const v2f*)(ap + k + 2 * half);
        v2f a1 = *(const v2f*)(ap + k + 4 + 2 * half);
        const float* bp0 = sW + (k / 2 + half) * 256 + 2 * l16;
        const float* bp1 = bp0 + 2 * 256;
        #pragma unroll
        for (int n = 0; n < 8; ++n) {
            v2f b0 = *(const v2f*)(bp0 + 32 * n);     // ds_load_b64
            acc[n] = wmma4(a0, b0, acc[n]);
        }
        #pragma unroll
        for (int n = 0; n < 8; ++n) {
            v2f b1 = *(const v2f*)(bp1 + 32 * n);     // ds_load_b64
            acc[n] = wmma4(a1, b1, acc[n]);
        }
    }

    #pragma unroll
    for (int n = 0; n < 8; ++n) {
        const int col = n * 16 + l16;
        const float bv = bias[col];
        #pragma unroll
        for (int v = 0; v < 8; ++v) {
            const int row = row0 + v + 8 * half;
            if (row < M) {
                float val = acc[n][v] + bv;
                if (EXTRA) val += extra[(size_t)extraIdx[row] * HH + col];
                if (RELU)  val = fmaxf(val, 0.f);
                out[(size_t)row * HH + col] = val;
            }
        }
    }
}

// ---------------------------------------------------------------------------
// aggr[dst] += relu(x[src] + attr*We + be)   (one wave per edge)
// We/be staged to LDS with gfx1250 async global->LDS DMA (ASYNCcnt-tracked).
// ---------------------------------------------------------------------------
__global__ __launch_bounds__(256)
void k_edge_aggr(const float* __restrict__ x, const int* __restrict__ ei,
                 const float* __restrict__ eattr,
                 const float* __restrict__ We, const float* __restrict__ be,
                 float* __restrict__ aggr) {
    __shared__ float sWe[HH];
    __shared__ float sbe[HH];
    if (threadIdx.x < 64) {
        const int t = threadIdx.x & 31;
        const float* g = (threadIdx.x < 32) ? (We + t * 4) : (be + t * 4);
        float*       l = (threadIdx.x < 32) ? (sWe + t * 4) : (sbe + t * 4);
        uint32_t laddr = (uint32_t)(uintptr_t)l;   // low 32 bits = LDS offset
        asm volatile("global_load_async_to_lds_b128 %0, %1, off"
                     :: "v"(laddr), "v"(g) : "memory");
    }
    asm volatile("s_wait_asynccnt 0x0" ::: "memory");
    __syncthreads();

    const int e = blockIdx.x * 8 + (threadIdx.x >> 5);
    if (e >= EE) return;
    const int lane = threadIdx.x & 31;
    const int src = ei[e];
    const int dst = ei[EE + e];
    const float a = eattr[e];
    const float* __restrict__ xs = x + (size_t)src * HH;
    float* __restrict__ ad = aggr + (size_t)dst * HH;
    __builtin_prefetch(ad, 1, 1);
    #pragma unroll
    for (int r = 0; r < 4; ++r) {
        const int j = lane + r * 32;
        float m = xs[j] + a * sWe[j] + sbe[j];
        m = fmaxf(m, 0.f);
        atomicAdd(&ad[j], m);                    // global_atomic_add_f32
    }
}

// ---------------------------------------------------------------------------
// small utility kernels
// ---------------------------------------------------------------------------
__global__ void k_copy4(float4* __restrict__ dst, const float4* __restrict__ src,
                        long n4) {
    long i = (long)blockIdx.x * blockDim.x + threadIdx.x;
    if (i < n4) dst[i] = src[i];
}

__global__ void k_zero(float* __restrict__ p, long n) {
    long i = (long)blockIdx.x * blockDim.x + threadIdx.x;
    if (i < n) p[i] = 0.f;
}

// zsum[c] += x[pidx[p]]; zcnt[c] += 1   (one wave per pool entry)
__global__ __launch_bounds__(256)
void k_pool_scatter(const float* __restrict__ x, const int* __restrict__ pidx,
                    const int* __restrict__ pmem, float* __restrict__ zsum,
                    float* __restrict__ zcnt) {
    const int p = blockIdx.x * 8 + (threadIdx.x >> 5);
    if (p >= PP) return;
    const int lane = threadIdx.x & 31;
    const int n = pidx[p];
    const int c = pmem[p];
    const float* __restrict__ xs = x + (size_t)n * HH;
    float* __restrict__ zs = zsum + (size_t)c * HH;
    #pragma unroll
    for (int r = 0; r < 4; ++r) {
        const int j = lane + r * 32;
        atomicAdd(&zs[j], xs[j]);
    }
    if (lane == 0) atomicAdd(&zcnt[c], 1.f);
}

__global__ void k_seg_div(float* __restrict__ v, const float* __restrict__ cnt,
                          long n) {
    long i = (long)blockIdx.x * blockDim.x + threadIdx.x;
    if (i < n) v[i] = v[i] / fmaxf(cnt[i / HH], 1.f);
}

// ctxsum[batch[c]] += z[c]; bcnt += 1  (one wave per candidate)
__global__ __launch_bounds__(256)
void k_ctx_scatter(const float* __restrict__ z, const int* __restrict__ cbatch,
                   float* __restrict__ ctxsum, float* __restrict__ bcnt) {
    const int c = blockIdx.x * 8 + (threadIdx.x >> 5);
    if (c >= CC) return;
    const int lane = threadIdx.x & 31;
    const int b = cbatch[c];
    const float* __restrict__ zs = z + (size_t)c * HH;
    float* __restrict__ cs = ctxsum + (size_t)b * HH;
    #pragma unroll
    for (int r = 0; r < 4; ++r) {
        const int j = lane + r * 32;
        atomicAdd(&cs[j], zs[j]);
    }
    if (lane == 0) atomicAdd(&bcnt[b], 1.f);
}

// ctxproj[b] = ctx[b] @ Wp1[128:256, :]   (only 16 distinct context rows)
__global__ void k_ctxproj(const float* __restrict__ ctx,
                          const float* __restrict__ Wp1,
                          float* __restrict__ ctxproj) {
    const int b = blockIdx.x;        // 16 blocks
    const int o = threadIdx.x;       // 128 threads
    float s = 0.f;
    for (int k = 0; k < HH; ++k)
        s += ctx[b * HH + k] * Wp1[(HH + k) * HH + o];
    ctxproj[b * HH + o] = s;
}

// logits = mask ? h2 @ Wp3 + bp3 : -1e9
__global__ void k_head_out(const float* __restrict__ h2,
                           const float* __restrict__ Wp3,
                           const float* __restrict__ bp3,
                           const unsigned char* __restrict__ mask,
                           float* __restrict__ out) {
    const int c = blockIdx.x * blockDim.x + threadIdx.x;
    if (c >= CC) return;
    const float4* __restrict__ hv = (const float4*)(h2 + (size_t)c * HH);
    const float4* __restrict__ wv = (const float4*)Wp3;
    float s = 0.f;
    #pragma unroll
    for (int j = 0; j < HH / 4; ++j) {
        float4 a = hv[j], b = wv[j];
        s += a.x * b.x + a.y * b.y + a.z * b.z + a.w * b.w;
    }
    out[c] = mask[c] ? (s + bp3[0]) : -1e9f;
}

// ---------------------------------------------------------------------------
extern "C" void kernel_launch(void* const* d_in, const int* in_sizes, int n_in,
                              void* d_out, int out_size, void* d_ws, size_t ws_size,
                              hipStream_t stream) {
    const float* nf     = (const float*)d_in[0];
    const int*   ei     = (const int*)  d_in[1];
    const float* eattr  = (const float*)d_in[2];
    // d_in[3] = membership (unused by reference)
    const int*   cbatch = (const int*)  d_in[4];
    const unsigned char* mask = (const unsigned char*)d_in[5];
    const int*   pidx   = (const int*)  d_in[6];
    const int*   pmem   = (const int*)  d_in[7];
    const float* Wi  = (const float*)d_in[8];
    const float* bi  = (const float*)d_in[9];
    const float* We  = (const float*)d_in[10];
    const float* be  = (const float*)d_in[11];
    const float* W1s = (const float*)d_in[12];
    const float* b1s = (const float*)d_in[13];
    const float* W2s = (const float*)d_in[14];
    const float* b2s = (const float*)d_in[15];
    const float* Wp1 = (const float*)d_in[16];
    const float* bp1 = (const float*)d_in[17];
    const float* Wp2 = (const float*)d_in[18];
    const float* bp2 = (const float*)d_in[19];
    const float* Wp3 = (const float*)d_in[20];
    const float* bp3 = (const float*)d_in[21];
    float* out = (float*)d_out;

    // workspace layout (floats)
    float* buf0    = (float*)d_ws;                      // [NN, 128]
    float* buf1    = buf0 + (size_t)NN * HH;            // [NN, 128]
    float* z       = buf1 + (size_t)NN * HH;            // [CC, 128]
    float* zcnt    = z + (size_t)CC * HH;               // [CC]
    float* ctxsum  = zcnt + CC;                         // [BB, 128]
    float* bcnt    = ctxsum + (size_t)BB * HH;          // [BB]
    float* ctx     = bcnt + BB;                         // [BB, 128]
    float* ctxproj = ctx + (size_t)BB * HH;             // [BB, 128]

    const dim3 blk(256);
    const int nodeBlocks = (NN + 127) / 128;            // 782

    // 1) x = node_features @ Wi + bi
    k_gemm<DIN, false, false><<<nodeBlocks, blk, 0, stream>>>(
        nf, Wi, bi, buf0, NN, nullptr, nullptr);

    // 2) GINE layers (ping-pong buf0/buf1)
    float* cur = buf0;
    float* alt = buf1;
    const long n4 = (long)NN * HH / 4;
    for (int l = 0; l < NLAYER; ++l) {
        k_copy4<<<(int)((n4 + 255) / 256), blk, 0, stream>>>(
            (float4*)alt, (const float4*)cur, n4);                 // aggr = x
        k_edge_aggr<<<(EE + 7) / 8, blk, 0, stream>>>(
            cur, ei, eattr, We, be, alt);                          // += messages
        k_gemm<HH, true, false><<<nodeBlocks, blk, 0, stream>>>(
            alt, W1s + (size_t)l * HH * HH, b1s + l * HH, cur, NN, nullptr, nullptr);
        k_gemm<HH, true, false><<<nodeBlocks, blk, 0, stream>>>(
            cur, W2s + (size_t)l * HH * HH, b2s + l * HH, alt, NN, nullptr, nullptr);
        float* t = cur; cur = alt; alt = t;                        // x = alt
    }
    // final node features in `cur`

    // 3) candidate pooling (mean)
    const long zeroN = (long)CC * HH + CC + (long)BB * HH + BB;
    k_zero<<<(int)((zeroN + 255) / 256), blk, 0, stream>>>(z, zeroN);
    k_pool_scatter<<<PP / 8, blk, 0, stream>>>(cur, pidx, pmem, z, zcnt);
    k_seg_div<<<(int)(((long)CC * HH + 255) / 256), blk, 0, stream>>>(
        z, zcnt, (long)CC * HH);

    // 4) global context (mean per graph), then project through Wp1[128:256]
    k_ctx_scatter<<<CC / 8, blk, 0, stream>>>(z, cbatch, ctxsum, bcnt);
    k_copy4<<<(BB * HH / 4 + 255) / 256, blk, 0, stream>>>(
        (float4*)ctx, (const float4*)ctxsum, BB * HH / 4);
    k_seg_div<<<(BB * HH + 255) / 256, blk, 0, stream>>>(ctx, bcnt, BB * HH);
    k_ctxproj<<<BB, HH, 0, stream>>>(ctx, Wp1, ctxproj);

    // 5) head MLP: h1 = relu(z@Wp1[0:128] + ctxproj[batch] + bp1)
    float* h1 = alt;                         // reuse big buffer not holding x
    float* h2 = alt + (size_t)CC * HH;
    const int headBlocks = (CC + 127) / 128; // 64
    k_gemm<HH, true, true><<<headBlocks, blk, 0, stream>>>(
        z, Wp1, bp1, h1, CC, ctxproj, cbatch);
    k_gemm<HH, true, false><<<headBlocks, blk, 0, stream>>>(
        h1, Wp2, bp2, h2, CC, nullptr, nullptr);
    k_head_out<<<(CC + 255) / 256, blk, 0, stream>>>(h2, Wp3, bp3, mask, out);
}